// EmbeddingModule_75213467287608
// MI455X (gfx1250) — compile-verified
//
#include <hip/hip_runtime.h>
#include <math.h>

typedef float v2f __attribute__((ext_vector_type(2)));
typedef float v4f __attribute__((ext_vector_type(4)));
typedef float v8f __attribute__((ext_vector_type(8)));

#define ROWS_PER_BLOCK 128
#define THREADS 128
#define PRE_LD 136   // 135 real K + 1 zero pad; row stride in LDS
#define H_LD 72      // 64 + pad

// d_in order (setup_inputs dict order):
// 0 dense(B,64) f32
// 1..6 ids_{rating,category,fandom,relationship,character,freeform} int32
// 7 W_dense(64,20)  8 b_dense(20)
// 9..14 tab_{rating,category,fandom,relationship,character,freeform}
// 15 W1(135,64) 16 b1(64) 17 W2(64,128) 18 b2(128)

template<int DIM, int LEN, int UNROLL>
__device__ __forceinline__ void pool_to(float* dst, const float* __restrict__ tab,
                                        const int* __restrict__ ids, int row) {
  float acc[DIM];
#pragma unroll
  for (int d = 0; d < DIM; ++d) acc[d] = 0.f;
#pragma unroll UNROLL
  for (int j = 0; j < LEN; ++j) {
    const int id = ids[row * LEN + j];
    float v[DIM];
    if constexpr ((DIM % 4) == 0) {
      // row = DIM*4 bytes, multiple of 16 -> 16B aligned: b128 gathers
      const v4f* tr = (const v4f*)(tab + (size_t)id * DIM);
#pragma unroll
      for (int c = 0; c < DIM / 4; ++c) {
        v4f t = tr[c];
        v[4 * c + 0] = t.x; v[4 * c + 1] = t.y;
        v[4 * c + 2] = t.z; v[4 * c + 3] = t.w;
      }
    } else {
      // row = 40 bytes, multiple of 8 -> 8B aligned: b64 gathers
      const v2f* tr = (const v2f*)(tab + (size_t)id * DIM);
#pragma unroll
      for (int c = 0; c < DIM / 2; ++c) {
        v2f t = tr[c];
        v[2 * c + 0] = t.x; v[2 * c + 1] = t.y;
      }
    }
    float ss = 0.f;
#pragma unroll
    for (int d = 0; d < DIM; ++d) ss = fmaf(v[d], v[d], ss);
    float n = sqrtf(ss);
    float s = fminf(1.0f, 1.0f / (n + 1e-7f));
#pragma unroll
    for (int d = 0; d < DIM; ++d) acc[d] = fmaf(v[d], s, acc[d]);
  }
#pragma unroll
  for (int d = 0; d < DIM; ++d) dst[d] = acc[d];
}

__device__ __forceinline__ v8f wmma_f32_k4(v2f a, v2f b, v8f c) {
#if __has_builtin(__builtin_amdgcn_wmma_f32_16x16x4_f32)
  // 8 args: (neg_a, A, neg_b, B, c_mod, C, reuse_a, reuse_b)
  return __builtin_amdgcn_wmma_f32_16x16x4_f32(false, a, false, b, (short)0, c,
                                               false, false);
#else
  asm volatile("v_wmma_f32_16x16x4_f32 %0, %1, %2, %0"
               : "+v"(c) : "v"(a), "v"(b));
  return c;
#endif
}

// Opaque copy of a uniform pointer: blocks LICM/CSE from hoisting the weight
// loads out of the tile loops (which previously forced scratch spills under
// the occupancy cap). Costs nothing at runtime.
__device__ __forceinline__ const float* launder(const float* p) {
  asm("" : "+s"(p));
  return p;
}

__global__ __launch_bounds__(THREADS)
__attribute__((amdgpu_waves_per_eu(4)))
void fused_embmlp_kernel(
    const float* __restrict__ dense,
    const int* __restrict__ ids_rating, const int* __restrict__ ids_category,
    const int* __restrict__ ids_fandom, const int* __restrict__ ids_relationship,
    const int* __restrict__ ids_character, const int* __restrict__ ids_freeform,
    const float* __restrict__ W_dense, const float* __restrict__ b_dense,
    const float* __restrict__ tab_rating, const float* __restrict__ tab_category,
    const float* __restrict__ tab_fandom, const float* __restrict__ tab_relationship,
    const float* __restrict__ tab_character, const float* __restrict__ tab_freeform,
    const float* __restrict__ W1, const float* __restrict__ b1,
    const float* __restrict__ W2, const float* __restrict__ b2,
    float* __restrict__ out) {
  extern __shared__ float smem[];
  float* s_pre = smem;                           // [128][PRE_LD]
  float* s_h   = smem + ROWS_PER_BLOCK * PRE_LD; // [4 waves][16][H_LD]

  const int tid = threadIdx.x;
  const int row = blockIdx.x * ROWS_PER_BLOCK + tid;
  float* myrow = s_pre + tid * PRE_LD;

  // ---------- Phase 1: per-row features -> pre_over[136] in LDS ----------
  {
    const v4f* dr4 = (const v4f*)(dense + (size_t)row * 64);  // 256B-aligned row
    float acc[20];
#pragma unroll
    for (int j = 0; j < 20; ++j) acc[j] = b_dense[j];
#pragma unroll 2
    for (int kc = 0; kc < 16; ++kc) {
      v4f x = dr4[kc];
#pragma unroll
      for (int j = 0; j < 20; ++j) {
        acc[j] = fmaf(x.x, W_dense[(4 * kc + 0) * 20 + j], acc[j]);
        acc[j] = fmaf(x.y, W_dense[(4 * kc + 1) * 20 + j], acc[j]);
        acc[j] = fmaf(x.z, W_dense[(4 * kc + 2) * 20 + j], acc[j]);
        acc[j] = fmaf(x.w, W_dense[(4 * kc + 3) * 20 + j], acc[j]);
      }
    }
#pragma unroll
    for (int j = 0; j < 20; ++j) myrow[j] = acc[j];
  }
  pool_to<10, 1, 1>(myrow + 20,  tab_rating,       ids_rating,       row);
  pool_to<10, 2, 2>(myrow + 30,  tab_category,     ids_category,     row);
  pool_to<20, 3, 2>(myrow + 40,  tab_fandom,       ids_fandom,       row);
  pool_to<20, 3, 2>(myrow + 60,  tab_relationship, ids_relationship, row);
  pool_to<20, 5, 2>(myrow + 80,  tab_character,    ids_character,    row);
  pool_to<20, 10, 2>(myrow + 100, tab_freeform,    ids_freeform,     row);

  {
    // vectors live contiguously at myrow[20*i .. 20*i+19]:
    // 0:de 1:rating||category 2:fandom 3:relationship 4:character 5:freeform
    const int PI[15] = {0,0,0,0,0,1,1,1,1,2,2,2,3,3,4};
    const int PJ[15] = {1,2,3,4,5,2,3,4,5,3,4,5,4,5,5};
#pragma unroll 3
    for (int p = 0; p < 15; ++p) {
      const float* a = myrow + 20 * PI[p];
      const float* b = myrow + 20 * PJ[p];
      float s = 0.f;
#pragma unroll
      for (int d = 0; d < 20; ++d) s = fmaf(a[d], b[d], s);
      myrow[120 + p] = s;
    }
    myrow[135] = 0.f;  // K pad
  }

  __syncthreads();

  // ---------- Phase 2: fp32 WMMA MLP (each wave owns the rows it built) ----
  const int wave = tid >> 5;      // wave32
  const int lane = tid & 31;
  const int lo = lane & 15;       // N (B/C/D) or M (A)
  const int hi = lane >> 4;       // selects K pair (A/B) or M+8 (C/D)
  float* hbase = s_h + wave * 16 * H_LD;

  for (int t = 0; t < 2; ++t) {
    const int m0 = (wave * 2 + t) * 16;

    // ---- MLP1: pre_over(16x136) @ W1(135x64) + b1 -> leaky -> s_h ----
    const float* arow = s_pre + (m0 + lo) * PRE_LD + 2 * hi;
#pragma unroll
    for (int nt = 0; nt < 4; ++nt) {
      const float* W1l = launder(W1);   // keep weight loads inside this iter
      const int n0 = nt * 16;
      const float bias = b1[n0 + lo];
      v8f c;
#pragma unroll
      for (int i = 0; i < 8; ++i) c[i] = bias;
#pragma unroll
      for (int k0 = 0; k0 < 136; k0 += 4) {
        v2f a; a.x = arow[k0]; a.y = arow[k0 + 1];
        const int kk = k0 + 2 * hi;
        v2f b;
        b.x = W1l[kk * 64 + n0 + lo];
        b.y = (kk + 1 < 135) ? W1l[(kk + 1) * 64 + n0 + lo] : 0.f;  // guard pad
        c = wmma_f32_k4(a, b, c);
      }
#pragma unroll
      for (int i = 0; i < 8; ++i) {
        float v = c[i];
        v = v > 0.f ? v : 0.01f * v;               // leaky relu
        hbase[(i + 8 * hi) * H_LD + n0 + lo] = v;  // M = i + 8*hi, N = lo
      }
    }
    asm volatile("s_wait_dscnt 0x0" ::: "memory");  // h stores visible to own wave

    // ---- MLP2: h(16x64) @ W2(64x128) + b2 -> out ----
    const float* hrow = hbase + lo * H_LD + 2 * hi;
    const size_t rbase = (size_t)(blockIdx.x * ROWS_PER_BLOCK + m0) * 128;
#pragma unroll
    for (int nt = 0; nt < 8; ++nt) {
      const float* W2l = launder(W2);   // keep weight loads inside this iter
      const int n0 = nt * 16;
      const float bias = b2[n0 + lo];
      v8f c;
#pragma unroll
      for (int i = 0; i < 8; ++i) c[i] = bias;
#pragma unroll
      for (int k0 = 0; k0 < 64; k0 += 4) {
        v2f a; a.x = hrow[k0]; a.y = hrow[k0 + 1];
        const int kk = k0 + 2 * hi;
        v2f b;
        b.x = W2l[kk * 128 + n0 + lo];
        b.y = W2l[(kk + 1) * 128 + n0 + lo];
        c = wmma_f32_k4(a, b, c);
      }
#pragma unroll
      for (int i = 0; i < 8; ++i)
        out[rbase + (size_t)(i + 8 * hi) * 128 + n0 + lo] = c[i];
    }
  }
}

extern "C" void kernel_launch(void* const* d_in, const int* in_sizes, int n_in,
                              void* d_out, int out_size, void* d_ws, size_t ws_size,
                              hipStream_t stream) {
  const float* dense = (const float*)d_in[0];
  const int* ids_rating       = (const int*)d_in[1];
  const int* ids_category     = (const int*)d_in[2];
  const int* ids_fandom       = (const int*)d_in[3];
  const int* ids_relationship = (const int*)d_in[4];
  const int* ids_character    = (const int*)d_in[5];
  const int* ids_freeform     = (const int*)d_in[6];
  const float* W_dense = (const float*)d_in[7];
  const float* b_dense = (const float*)d_in[8];
  const float* tab_rating       = (const float*)d_in[9];
  const float* tab_category     = (const float*)d_in[10];
  const float* tab_fandom       = (const float*)d_in[11];
  const float* tab_relationship = (const float*)d_in[12];
  const float* tab_character    = (const float*)d_in[13];
  const float* tab_freeform     = (const float*)d_in[14];
  const float* W1 = (const float*)d_in[15];
  const float* b1 = (const float*)d_in[16];
  const float* W2 = (const float*)d_in[17];
  const float* b2 = (const float*)d_in[18];
  float* out = (float*)d_out;

  const int Btot = in_sizes[0] / 64;              // 65536
  const int grid = Btot / ROWS_PER_BLOCK;         // 512
  const size_t shmem = (size_t)(ROWS_PER_BLOCK * PRE_LD + 4 * 16 * H_LD) * sizeof(float);

  fused_embmlp_kernel<<<grid, THREADS, shmem, stream>>>(
      dense, ids_rating, ids_category, ids_fandom, ids_relationship,
      ids_character, ids_freeform, W_dense, b_dense, tab_rating, tab_category,
      tab_fandom, tab_relationship, tab_character, tab_freeform, W1, b1, W2, b2,
      out);
}